// RankRTModelA_52192442581569
// MI455X (gfx1250) — compile-verified
//
#include <hip/hip_runtime.h>
#include <stdint.h>

#define BLOCK 256
#define ROWS  4   // rows per thread -> 80B of indices (5x int4), 64B of p, 16B of rt

typedef __attribute__((ext_vector_type(4))) unsigned int u32x4;
typedef __attribute__((ext_vector_type(4))) int          i32x4;
typedef __attribute__((ext_vector_type(8))) int          i32x8;

// Single shared body for the (never-taken-in-practice) general-rho/tau path.
// noinline keeps ~80 pow expansions from being inlined into the kernel.
__device__ __noinline__ float slow_pow(float x, float y) {
    return __powf(x, y);
}

__global__ __launch_bounds__(BLOCK) void rankrt_kernel(
    const int*   __restrict__ gidx,     // (B,5) int32, values in [1,20]
    const float* __restrict__ percept,  // (21,3) f32
    const float* __restrict__ w,        // (3,)
    const float* __restrict__ prho, const float* __restrict__ pbeta,
    const float* __restrict__ ptau, const float* __restrict__ pgamma,
    const float* __restrict__ pupper, const float* __restrict__ pmid,
    const float* __restrict__ prate,
    float* __restrict__ outp,           // (B,4)
    float* __restrict__ outrt,          // (B,)
    int B)
{
    __shared__ float sper[64];          // 21*3 = 63 floats, padded to 64
    const int tid = threadIdx.x;

    // ---- Stage percept table into LDS via the Tensor Data Mover ----
#if __has_builtin(__builtin_amdgcn_tensor_load_to_lds)
    if (tid == 0) {
        const unsigned long long ga = (unsigned long long)(uintptr_t)percept;
        const unsigned int lds_addr = (unsigned int)(uintptr_t)(&sper[0]);
        // D# group 0: count=1, lds_addr, global_addr[56:0], type=2 ("image")
        u32x4 g0;
        g0[0] = 1u;                                       // count = 1 valid descriptor
        g0[1] = lds_addr;                                 // LDS byte address
        g0[2] = (unsigned int)(ga & 0xFFFFFFFFull);       // global_addr[31:0]
        g0[3] = (unsigned int)((ga >> 32) & 0x01FFFFFFull)// global_addr[56:32]
              | 0x80000000u;                              // type = 2 at bits[127:126]
        // D# group 1: data_size=4B; 2D tensor 63x1; tile 63x1; stride0=63
        i32x8 g1 = {};
        g1[0] = (int)(2u << 16);      // data_size = 2 -> 4 bytes (bits 17:16)
        g1[1] = (int)(63u << 16);     // tensor_dim0[15:0]  (bits 63:48)
        g1[2] = (int)(1u << 16);      // tensor_dim1[15:0]  (bits 95:80)
        g1[3] = (int)(63u << 16);     // tile_dim0          (bits 127:112)
        g1[4] = 1;                    // tile_dim1 = 1      (bits 143:128)
        g1[5] = 63;                   // tensor_dim0_stride (bits 191:160 low)
        i32x4 g2 = {};                // dims 2/3 unused
        i32x4 g3 = {};
#if defined(__clang_major__) && __clang_major__ >= 23
        i32x8 g4 = {};
        __builtin_amdgcn_tensor_load_to_lds(g0, g1, g2, g3, g4, 0);
#else
        __builtin_amdgcn_tensor_load_to_lds(g0, g1, g2, g3, 0);
#endif
        __builtin_amdgcn_s_wait_tensorcnt(0);
    }
#else
    if (tid < 63) sper[tid] = percept[tid];
#endif
    __syncthreads();

    // ---- Uniform scalars (SGPR-resident) ----
    const float w0 = w[0], w1 = w[1], w2 = w[2];
    const float rho  = prho[0],  beta = pbeta[0], tau  = ptau[0];
    const float gma  = pgamma[0], upper = pupper[0];
    const float mid  = pmid[0],  rate = prate[0];
    const float inv_rho = 1.0f / rho;
    const bool  fast    = (rho == 2.0f) & (tau == 1.0f); // uniform -> SALU branch

    const long long row0 = (long long)blockIdx.x * (BLOCK * ROWS)
                         + (long long)tid * ROWS;
    if (row0 >= B) return;

    // ---- 80 bytes of indices as 5x global_load_b128 (16B aligned) ----
    int ind[ROWS * 5];
    {
        const int4* src = (const int4*)(gidx + row0 * 5);
        int4* dst = (int4*)ind;
#pragma unroll
        for (int i = 0; i < 5; ++i) dst[i] = src[i];
    }

    float4 pout[ROWS];
    float  rts[ROWS];
#pragma unroll
    for (int r = 0; r < ROWS; ++r) {
        const int* ri = &ind[r * 5];
        const int q = ri[0] * 3;
        const float zq0 = sper[q], zq1 = sper[q + 1], zq2 = sper[q + 2];

        float sv[4];
        float ssum = 0.0f;
#pragma unroll
        for (int j = 0; j < 4; ++j) {
            const int b = ri[j + 1] * 3;
            const float a0 = fabsf(zq0 - sper[b]);
            const float a1 = fabsf(zq1 - sper[b + 1]);
            const float a2 = fabsf(zq2 - sper[b + 2]);
            float dt;
            if (fast) {
                // rho==2, tau==1: d = sqrt(sum w*a^2), d^tau = d
                dt = __fsqrt_rn(w0 * a0 * a0 + w1 * a1 * a1 + w2 * a2 * a2);
            } else {
                const float acc = w0 * slow_pow(a0, rho)
                                + w1 * slow_pow(a1, rho)
                                + w2 * slow_pow(a2, rho);
                const float d = slow_pow(acc, inv_rho);
                dt = slow_pow(d, tau);
            }
            const float v = __expf(-beta * dt) + gma;
            sv[j] = v;
            ssum += v;
        }
        const float inv = 1.0f / ssum;
        const float p0 = sv[0] * inv, p1 = sv[1] * inv;
        const float p2 = sv[2] * inv, p3 = sv[3] * inv;
        const float ent = -(p0 * __logf(p0) + p1 * __logf(p1)
                          + p2 * __logf(p2) + p3 * __logf(p3));
        pout[r] = make_float4(p0, p1, p2, p3);
        rts[r]  = upper / (1.0f + __expf(-rate * (ent - mid)));
    }

    // ---- 64B of p as 4x global_store_b128, 16B of rt as 1x b128 ----
    float4* op = (float4*)(outp + row0 * 4);
#pragma unroll
    for (int r = 0; r < ROWS; ++r) op[r] = pout[r];
    *((float4*)(outrt + row0)) = make_float4(rts[0], rts[1], rts[2], rts[3]);
}

extern "C" void kernel_launch(void* const* d_in, const int* in_sizes, int n_in,
                              void* d_out, int out_size, void* d_ws, size_t ws_size,
                              hipStream_t stream) {
    const int B = in_sizes[0] / 5;

    const int*   gidx    = (const int*)  d_in[0];
    const float* percept = (const float*)d_in[1];
    const float* w       = (const float*)d_in[2];
    const float* rho     = (const float*)d_in[3];
    const float* beta    = (const float*)d_in[4];
    const float* tau     = (const float*)d_in[5];
    const float* gamma_  = (const float*)d_in[6];
    const float* upper   = (const float*)d_in[7];
    const float* mid     = (const float*)d_in[8];
    const float* rate    = (const float*)d_in[9];

    float* outp  = (float*)d_out;
    float* outrt = outp + (size_t)B * 4;

    const int rows_per_block = BLOCK * ROWS;                 // 1024
    const int grid = (B + rows_per_block - 1) / rows_per_block;

    rankrt_kernel<<<grid, BLOCK, 0, stream>>>(
        gidx, percept, w, rho, beta, tau, gamma_, upper, mid, rate,
        outp, outrt, B);
}